// CPUConv2D_54709293416733
// MI455X (gfx1250) — compile-verified
//
#include <hip/hip_runtime.h>

typedef __attribute__((ext_vector_type(2))) float v2f;
typedef __attribute__((ext_vector_type(8))) float v8f;

#define C_CH   384
#define HW     512
#define KH_    7
#define KW_    7
#define PADR   3
#define TW     128        // output cols per workgroup (8 waves x 16)
#define TH     64         // output rows per workgroup (4 row-tiles of 16)
#define RT     4          // row-tiles per wave
#define XS_ROWS   70      // TH + KH - 1
#define XS_STRIDE 138     // TW + 6 = 134, padded to 138: stride%64 = 10 -> conflict-free A rows
#define WPAD   40         // zero-padded taps per kh: index t+16, t in [-16,24)

#if __has_builtin(__builtin_amdgcn_global_load_async_to_lds_b32)
#define HAVE_ASYNC_LDS 1
#else
#define HAVE_ASYNC_LDS 0
#endif

typedef __attribute__((address_space(1))) int g_int;
typedef __attribute__((address_space(3))) int l_int;

__device__ __forceinline__ void wait_asynccnt0() {
#if HAVE_ASYNC_LDS
#if __has_builtin(__builtin_amdgcn_s_wait_asynccnt)
  __builtin_amdgcn_s_wait_asynccnt(0);
#else
  asm volatile("s_wait_asynccnt 0x0" ::: "memory");
#endif
#endif
}

__launch_bounds__(256)
__global__ void dwconv7x7_wmma_f32(const float* __restrict__ X,
                                   const float* __restrict__ Wt,
                                   float* __restrict__ Y) {
  __shared__ float Xs[XS_ROWS * XS_STRIDE];   // 38,640 B
  __shared__ float Wp[KH_ * WPAD];            //  1,120 B

  const int ch   = blockIdx.z;
  const int row0 = blockIdx.y * TH;
  const int col0 = blockIdx.x * TW;
  const int tid  = threadIdx.x;

  const float* Xc = X + (size_t)ch * HW * HW;
  const float* Wc = Wt + (size_t)ch * (KH_ * KW_);

  // Stage zero-padded weights: Wp[kh*WPAD + (t+16)] = w[kh][t] for 0<=t<7, else 0.
  for (int i = tid; i < KH_ * WPAD; i += 256) {
    int kh = i / WPAD;
    int t  = (i % WPAD) - 16;
    Wp[i] = (t >= 0 && t < KW_) ? Wc[kh * KW_ + t] : 0.0f;
  }

  // Stage input tile (incl. halo). In-bounds cells go through the gfx1250
  // async global->LDS path (ASYNCcnt, no VGPR round-trip); out-of-bounds
  // cells are explicitly zeroed (WMMA propagates NaN, so no cell may be
  // left uninitialized).
  for (int i = tid; i < XS_ROWS * XS_STRIDE; i += 256) {
    int rr = i / XS_STRIDE;
    int cc = i % XS_STRIDE;
    int r  = row0 - PADR + rr;
    int c  = col0 - PADR + cc;
    bool inb = (r >= 0) && (r < HW) && (c >= 0) && (c < HW);
#if HAVE_ASYNC_LDS
    if (inb) {
      __builtin_amdgcn_global_load_async_to_lds_b32(
          (g_int*)(Xc + (size_t)r * HW + c),
          (l_int*)&Xs[i],
          /*offset=*/0, /*cpol=*/0);
    } else {
      Xs[i] = 0.0f;
    }
#else
    Xs[i] = inb ? Xc[(size_t)r * HW + c] : 0.0f;
#endif
  }
  wait_asynccnt0();
  __syncthreads();

  // Wave-level 16x16 output tiles via V_WMMA_F32_16X16X4_F32 (Toeplitz-B).
  const int wave = tid >> 5;   // 0..7 : column sub-tile
  const int lane = tid & 31;
  const int l16  = lane & 15;  // A row m / B col n / D col n
  const int half = lane >> 4;  // K-pair select: element e -> K = e + 2*half

  // Hoist all B fragments (depend only on kh, chunk) into registers:
  // B[k][n] = w[kh][k - n], k = 4*c4 + 2*half + e, n = l16.
  v2f bf[KH_][6];
#pragma unroll
  for (int kh = 0; kh < KH_; ++kh) {
    const float* wrow = &Wp[kh * WPAD + 16 + 2 * half - l16];
#pragma unroll
    for (int c4 = 0; c4 < 6; ++c4) {
      bf[kh][c4].x = wrow[4 * c4];
      bf[kh][c4].y = wrow[4 * c4 + 1];
    }
  }

  for (int rt = 0; rt < RT; ++rt) {
    v8f acc = {};
#pragma unroll
    for (int kh = 0; kh < KH_; ++kh) {
      // A base: Xs[rt*16 + m + kh][wave*16 + 2*half + k'], 8-byte aligned.
      const float* xrow =
          &Xs[(rt * 16 + l16 + kh) * XS_STRIDE + wave * 16 + 2 * half];
#pragma unroll
      for (int c4 = 0; c4 < 6; ++c4) {   // K = 24 (22 live + 2 zero-tap pad)
        v2f a = *(const v2f*)(xrow + 4 * c4);
        acc = __builtin_amdgcn_wmma_f32_16x16x4_f32(
            /*neg_a=*/false, a, /*neg_b=*/false, bf[kh][c4],
            /*c_mod=*/(short)0, acc, /*reuse_a=*/false, /*reuse_b=*/false);
      }
    }

    // D layout: VGPR v, lane L -> M = v + 8*(L>=16), N = L&15
    float* Yc = Y + (size_t)ch * HW * HW
              + (size_t)(row0 + rt * 16) * HW + col0 + wave * 16 + l16;
#pragma unroll
    for (int v = 0; v < 8; ++v) {
      int m = v + 8 * half;
      Yc[(size_t)m * HW] = acc[v];
    }
  }
}

extern "C" void kernel_launch(void* const* d_in, const int* in_sizes, int n_in,
                              void* d_out, int out_size, void* d_ws, size_t ws_size,
                              hipStream_t stream) {
  const float* X  = (const float*)d_in[0];   // (384, 512, 512) fp32
  const float* Wt = (const float*)d_in[1];   // (384, 1, 7, 7)  fp32
  float* Y = (float*)d_out;                  // (384, 512, 512) fp32

  dim3 grid(HW / TW, HW / TH, C_CH);         // (4, 8, 384)
  dwconv7x7_wmma_f32<<<grid, 256, 0, stream>>>(X, Wt, Y);
}